// YOLOLoss_22986664968619
// MI455X (gfx1250) — compile-verified
//
#include <hip/hip_runtime.h>

// ---------------------------------------------------------------------------
// YOLO loss reduction for MI455X (gfx1250, wave32).
// Bandwidth-bound single pass + exact-fp32 wave reduction via
// V_WMMA_F32_16X16X4_F32 (ones-matrix trick: D[m][n] = v[m] + v[m+16]).
// ---------------------------------------------------------------------------

typedef __attribute__((ext_vector_type(2))) float v2f;
typedef __attribute__((ext_vector_type(8))) float v8f;

#define NN 64
#define SSZ 128
#define BBOX 2
#define TOTAL_CELLS (NN * SSZ * SSZ)   // 1,048,576
#define NBLK 1024
#define TPB 256
#define NWAVE (TPB / 32)
#define NACC 6
#define WS_STRIDE 8                    // floats per block row in workspace
#define EPSI 1e-10f

__device__ __forceinline__ float sigmoid_f(float x) {
    return 1.0f / (1.0f + __expf(-x));
}

// Numerically stable log(1 + exp(x))  ==  jnp.logaddexp(0, x)
__device__ __forceinline__ float softplus_f(float x) {
    return fmaxf(x, 0.0f) + log1pf(__expf(-fabsf(x)));
}

// Exact fp32 wave32 sum using the f32 WMMA pipe.
// A = [v, 0] per lane (16x4), B = ones (4x16):
//   D[m][n] = v[m] + v[m+16]   for all n
// D layout: lane L<16 holds D[0..7][L] in its 8 regs -> sum = r[0..7];
//           lane L>=16 holds D[8..15][L-16]          -> sum = r[8..15].
// One shfl_xor(16) completes the 32-lane total on every lane.
// Requires EXEC == all ones (call only from fully converged code).
__device__ __forceinline__ float wave_sum32(float v) {
    v2f a; a[0] = v;    a[1] = 0.0f;
    v2f b; b[0] = 1.0f; b[1] = 1.0f;
    v8f c = {};
    v8f d = __builtin_amdgcn_wmma_f32_16x16x4_f32(
        /*neg_a=*/false, a, /*neg_b=*/false, b,
        /*c_mod=*/(short)0, c, /*reuse_a=*/false, /*reuse_b=*/false);
    float s = ((d[0] + d[1]) + (d[2] + d[3])) + ((d[4] + d[5]) + (d[6] + d[7]));
    s += __shfl_xor(s, 16, 32);
    return s;
}

__global__ __launch_bounds__(TPB) void yolo_partial_kernel(
    const float* __restrict__ input,   // [cells][10]
    const float* __restrict__ target,  // [cells][5]
    float* __restrict__ ws)            // [NBLK][WS_STRIDE]
{
    float am = 0.f, anoobj = 0.f, axy = 0.f, awh = 0.f, aobj = 0.f, anr = 0.f;

    const int stride = gridDim.x * blockDim.x;
    for (int cidx = blockIdx.x * blockDim.x + threadIdx.x;
         cidx < TOTAL_CELLS; cidx += stride) {

        // input cell: 10 floats, 40B stride -> 8B aligned -> b64 loads
        const float2* ip = reinterpret_cast<const float2*>(input) + (size_t)cidx * 5;
        float2 i0 = ip[0];   // (conf0, x0)
        float2 i1 = ip[1];   // (y0,    w0)
        float2 i2 = ip[2];   // (h0, conf1)
        float2 i3 = ip[3];   // (x1,    y1)
        float2 i4 = ip[4];   // (w1,    h1)

        const float* tp = target + (size_t)cidx * 5;
        float t0 = tp[0], tx = tp[1], ty = tp[2], tw = tp[3], th = tp[4];
        float obj = (t0 > 0.0f) ? 1.0f : 0.0f;

        float cl0 = i0.x, cl1 = i2.y;
        float sx0 = sigmoid_f(i0.y), sy0 = sigmoid_f(i1.x);
        float sw0 = sigmoid_f(i1.y), sh0 = sigmoid_f(i2.x);
        float sx1 = sigmoid_f(i3.x), sy1 = sigmoid_f(i3.y);
        float sw1 = sigmoid_f(i4.x), sh1 = sigmoid_f(i4.y);

        float bce0_0 = softplus_f(cl0);
        float bce0_1 = softplus_f(cl1);

        // target box corners & area (shared across both predicted boxes)
        float tlx = tx - 0.5f * tw, tly = ty - 0.5f * th;
        float trx = tx + 0.5f * tw, trb = ty + 0.5f * th;
        float area_t = tw * th;

        auto iou_of = [&](float sx, float sy, float sw, float sh) -> float {
            float wx = fminf(sx + 0.5f * sw, trx) - fmaxf(sx - 0.5f * sw, tlx);
            float wy = fminf(sy + 0.5f * sh, trb) - fmaxf(sy - 0.5f * sh, tly);
            wx = fmaxf(wx, 0.0f);
            wy = fmaxf(wy, 0.0f);
            float inter = wx * wy;
            return inter / (sw * sh + area_t - inter + EPSI);
        };

        float iou0 = iou_of(sx0, sy0, sw0, sh0);
        float iou1 = iou_of(sx1, sy1, sw1, sh1);

        // argmax over B=2 (ties -> box 0, matching jnp.argmax)
        bool r1 = (iou1 > iou0);
        float rx = r1 ? sx1 : sx0, ry = r1 ? sy1 : sy0;
        float rw = r1 ? sw1 : sw0, rh = r1 ? sh1 : sh0;
        float cl_r    = r1 ? cl1    : cl0;
        float bce0_r  = r1 ? bce0_1 : bce0_0;
        float bce0_nr = r1 ? bce0_0 : bce0_1;

        am     += obj;
        anoobj += (1.0f - obj) * (bce0_0 + bce0_1);
        float dx = rx - tx, dy = ry - ty, dw = rw - tw, dh = rh - th;
        axy  += obj * (dx * dx + dy * dy);
        awh  += obj * (dw * dw + dh * dh);
        aobj += obj * (bce0_r - cl_r);   // bce1 = bce0 - logit
        anr  += obj * bce0_nr;
    }

    // --- wave32 exact reduction via WMMA, then cross-wave via LDS ---
    __shared__ float smem[NWAVE][NACC];
    float acc[NACC] = {am, anoobj, axy, awh, aobj, anr};
    const int lane = threadIdx.x & 31;
    const int wave = threadIdx.x >> 5;
#pragma unroll
    for (int k = 0; k < NACC; ++k) {
        float s = wave_sum32(acc[k]);
        if (lane == 0) smem[wave][k] = s;
    }
    __syncthreads();
    if (threadIdx.x < NACC) {
        float s = 0.0f;
#pragma unroll
        for (int w = 0; w < NWAVE; ++w) s += smem[w][threadIdx.x];
        ws[(size_t)blockIdx.x * WS_STRIDE + threadIdx.x] = s;
    }
}

__global__ __launch_bounds__(TPB) void yolo_final_kernel(
    const float* __restrict__ ws, float* __restrict__ out)
{
    float p[NACC] = {0.f, 0.f, 0.f, 0.f, 0.f, 0.f};
    for (int i = threadIdx.x; i < NBLK; i += TPB) {
#pragma unroll
        for (int k = 0; k < NACC; ++k) p[k] += ws[(size_t)i * WS_STRIDE + k];
    }

    __shared__ float smem[NWAVE][NACC];
    const int lane = threadIdx.x & 31;
    const int wave = threadIdx.x >> 5;
#pragma unroll
    for (int k = 0; k < NACC; ++k) {
        float s = wave_sum32(p[k]);
        if (lane == 0) smem[wave][k] = s;
    }
    __syncthreads();

    if (threadIdx.x == 0) {
        float t[NACC];
#pragma unroll
        for (int k = 0; k < NACC; ++k) {
            float s = 0.0f;
#pragma unroll
            for (int w = 0; w < NWAVE; ++w) s += smem[w][k];
            t[k] = s;
        }
        float m       = t[0];
        float n_noobj = (float)TOTAL_CELLS - m;
        float loss_noobj = t[1] / (n_noobj * (float)BBOX) + t[5] / (m * (float)(BBOX - 1));
        float loss_xy    = t[2] / (2.0f * m);
        float loss_wh    = t[3] / (2.0f * m);
        float loss_obj   = t[4] / m;
        float loss = loss_noobj + loss_xy + loss_wh + loss_obj;
        out[0] = loss;
        out[1] = loss_noobj;
        out[2] = loss_xy;
        out[3] = loss_wh;
        out[4] = loss_obj;
    }
}

extern "C" void kernel_launch(void* const* d_in, const int* in_sizes, int n_in,
                              void* d_out, int out_size, void* d_ws, size_t ws_size,
                              hipStream_t stream) {
    const float* input  = (const float*)d_in[0];   // (64,128,128,10) f32
    const float* target = (const float*)d_in[1];   // (64,128,128,5)  f32
    float* ws  = (float*)d_ws;                     // NBLK * WS_STRIDE * 4 = 32 KB
    float* out = (float*)d_out;                    // 5 f32 scalars

    yolo_partial_kernel<<<NBLK, TPB, 0, stream>>>(input, target, ws);
    yolo_final_kernel<<<1, TPB, 0, stream>>>(ws, out);
}